// PerNeuronHead_4544075399232
// MI455X (gfx1250) — compile-verified
//
#include <hip/hip_runtime.h>
#include <hip/hip_bf16.h>
#include <cmath>

// Problem constants (match reference)
#define DIMc 256
#define EMBc 128
#define HIDc 128
#define Bc   4
#define NBc  512
#define NUc  512

typedef __attribute__((ext_vector_type(2))) float v2f;
typedef __attribute__((ext_vector_type(8))) float v8f;

// ---------------------------------------------------------------------------
// Kernel 1: out[M,HID] = A[M,K] @ W[K,HID]  (+ optional b1 broadcast)
// One wave32 -> one 16x16 output tile via V_WMMA_F32_16X16X4_F32.
// Block = 256 threads = 8 waves = one 16-row stripe covering HID=128 columns.
// ---------------------------------------------------------------------------
__global__ __launch_bounds__(256)
void proj_wmma_kernel(const float* __restrict__ A,
                      const float* __restrict__ W,
                      const float* __restrict__ b1,   // may be null
                      float* __restrict__ out,
                      int K, int addBias) {
  const int lane = threadIdx.x & 31;
  const int wave = threadIdx.x >> 5;        // 0..7 -> column tile
  const int tile_m = blockIdx.x * 16;
  const int tile_n = wave * 16;

  const int lo = lane & 15;                 // M (A) / N (B,C,D) within tile
  const int hi = lane >> 4;                 // half-wave select

  v8f c = {};                               // 16x16 f32 accumulator (8 VGPRs)
  const float* arow = A + (size_t)(tile_m + lo) * K;

  for (int kk = 0; kk < K; kk += 4) {
    // A 16x4 (ISA layout): lanes 0-15 K={kk,kk+1}, lanes 16-31 K={kk+2,kk+3}
    v2f a = *(const v2f*)(arow + kk + hi * 2);
    // B 4x16: VGPR0 = rows {kk,kk+1} striped by half-wave, VGPR1 = rows {kk+2,kk+3}
    v2f b;
    b.x = W[(size_t)(kk + hi)     * HIDc + tile_n + lo];
    b.y = W[(size_t)(kk + 2 + hi) * HIDc + tile_n + lo];
    // 8 args: (neg_a, A, neg_b, B, c_mod, C, reuse_a, reuse_b)
    c = __builtin_amdgcn_wmma_f32_16x16x4_f32(false, a, false, b,
                                              (short)0, c, false, false);
  }

  const float bias = addBias ? b1[tile_n + lo] : 0.0f;
  // C/D layout: VGPR r -> row (r + 8*hi), col = lo
  const int row0 = tile_m + hi * 8;
  const int col  = tile_n + lo;
#pragma unroll
  for (int r = 0; r < 8; ++r) {
    out[(size_t)(row0 + r) * HIDc + col] = c[r] + bias;
  }
}

// ---------------------------------------------------------------------------
// Kernel 2: fused broadcast-add + exact gelu + dot(W2) over h.
// Block = 256 threads -> one 16(n) x 32(u) output tile; each thread computes
// two outputs (u and u+16) so every lane carries two independent erf chains
// (the erf polynomial is a long dependent FMA chain -> ILP matters).
// LDS tiles use a 132-float padded stride: bank = (4*tx + h) mod 64, and the
// +16-row offset is bank-neutral (16*132 = 33*64), so all reads are
// conflict-free / broadcast.
// ---------------------------------------------------------------------------
__global__ __launch_bounds__(256)
void fused_gelu_dot_kernel(const float* __restrict__ binp,   // [B,NB,HID], b1 folded in
                           const float* __restrict__ unitp,  // [B,NU,HID]
                           const float* __restrict__ W2,     // [HID]
                           const float* __restrict__ b2,     // [1]
                           float* __restrict__ out) {        // [B,NB,NU]
  __shared__ float sB[16][132];
  __shared__ float sU[32][132];
  __shared__ float sW2[HIDc];

  const int b  = blockIdx.z;
  const int nt = blockIdx.y;                // 16-row n tile
  const int ut = blockIdx.x;                // 32-col u tile
  const int t  = threadIdx.x;

  const float* srcB = binp  + (size_t)(b * NBc + nt * 16) * HIDc;
  const float* srcU = unitp + (size_t)(b * NUc + ut * 32) * HIDc;

#pragma unroll
  for (int ch = 0; ch < 2; ++ch) {          // 2048 floats of binp tile
    const int lin = (ch * 256 + t) * 4;
    const int row = lin >> 7;
    const int col = lin & 127;
    *(float4*)&sB[row][col] = *(const float4*)(srcB + lin);
  }
#pragma unroll
  for (int ch = 0; ch < 4; ++ch) {          // 4096 floats of unitp tile
    const int lin = (ch * 256 + t) * 4;
    const int row = lin >> 7;
    const int col = lin & 127;
    *(float4*)&sU[row][col] = *(const float4*)(srcU + lin);
  }
  if (t < HIDc) sW2[t] = W2[t];
  __syncthreads();

  const int tx = t & 15;                    // u within tile (pair: tx, tx+16)
  const int ty = t >> 4;                    // n within tile (0..15)

  float acc0 = 0.0f;
  float acc1 = 0.0f;
#pragma unroll 4
  for (int h = 0; h < HIDc; ++h) {
    const float xa = sB[ty][h];
    const float w  = sW2[h];
    const float x0 = xa + sU[tx][h];
    const float x1 = xa + sU[tx + 16][h];
    const float g0 = 0.5f * x0 * (1.0f + erff(x0 * 0.70710678118654752f));
    const float g1 = 0.5f * x1 * (1.0f + erff(x1 * 0.70710678118654752f));
    acc0 = fmaf(g0, w, acc0);
    acc1 = fmaf(g1, w, acc1);
  }

  const int n = nt * 16 + ty;
  const int u = ut * 32 + tx;
  const float bias2 = b2[0];
  float* orow = out + (size_t)(b * NBc + n) * NUc;
  orow[u]      = acc0 + bias2;
  orow[u + 16] = acc1 + bias2;
}

// ---------------------------------------------------------------------------
extern "C" void kernel_launch(void* const* d_in, const int* in_sizes, int n_in,
                              void* d_out, int out_size, void* d_ws, size_t ws_size,
                              hipStream_t stream) {
  const float* bin_repr = (const float*)d_in[0];   // (4,512,256)
  const float* unit_emb = (const float*)d_in[1];   // (4,512,128)
  const float* W1       = (const float*)d_in[2];   // (384,128)
  const float* b1       = (const float*)d_in[3];   // (128)
  const float* W2       = (const float*)d_in[4];   // (128)
  const float* b2       = (const float*)d_in[5];   // (1)
  float* out = (float*)d_out;

  float* bin_proj  = (float*)d_ws;                          // 2048*128 f32
  float* unit_proj = bin_proj + (size_t)Bc * NBc * HIDc;    // 2048*128 f32

  const int Mrows = Bc * NBc;  // 2048

  // bin_proj = bin_repr @ W1[:256] + b1
  proj_wmma_kernel<<<Mrows / 16, 256, 0, stream>>>(
      bin_repr, W1, b1, bin_proj, DIMc, 1);
  // unit_proj = unit_emb @ W1[256:]
  proj_wmma_kernel<<<Mrows / 16, 256, 0, stream>>>(
      unit_emb, W1 + (size_t)DIMc * HIDc, nullptr, unit_proj, EMBc, 0);

  dim3 grid(NUc / 32, NBc / 16, Bc);
  fused_gelu_dot_kernel<<<grid, 256, 0, stream>>>(
      bin_proj, unit_proj, W2, b2, out);
}